// GIN_8486855377283
// MI455X (gfx1250) — compile-verified
//
#include <hip/hip_runtime.h>
#include <hip/hip_bf16.h>

typedef __attribute__((ext_vector_type(2))) float v2f;
typedef __attribute__((ext_vector_type(8))) float v8f;

#define DD 128
#define HROW 132   // padded LDS row pitch (floats) to avoid 64-bank conflicts

// ---------------------------------------------------------------------------
// Repack W[k][n] (row-major D x D, L layers) into pair-interleaved layout:
//   Wp[l][k/2][n] = { W[l][2*(k/2)][n], W[l][2*(k/2)+1][n] }
// so a WMMA B-fragment is one aligned b64 load per lane.
// ---------------------------------------------------------------------------
__global__ __launch_bounds__(256) void gin_repack(const float* __restrict__ W,
                                                  float* __restrict__ Wp,
                                                  int totalPairs) {
    int i = blockIdx.x * blockDim.x + threadIdx.x;   // i = l*(64*128) + kh*128 + n
    if (i >= totalPairs) return;
    int n  = i & 127;
    int kh = (i >> 7) & 63;
    int l  = i >> 13;
    const float* Wl = W + (size_t)l * DD * DD;
    float2 p;
    p.x = Wl[(2 * kh) * DD + n];
    p.y = Wl[(2 * kh + 1) * DD + n];
    ((float2*)Wp)[i] = p;
}

// ---------------------------------------------------------------------------
// z = (1 + eps[l]) * h      (vectorized float4)
// ---------------------------------------------------------------------------
__global__ __launch_bounds__(256) void gin_init_z(const float* __restrict__ h,
                                                  const float* __restrict__ eps,
                                                  int layer, float* __restrict__ z,
                                                  int n4) {
    int i = blockIdx.x * blockDim.x + threadIdx.x;
    if (i >= n4) return;
    float e = 1.0f + eps[layer];
    float4 v = ((const float4*)h)[i];
    v.x *= e; v.y *= e; v.z *= e; v.w *= e;
    ((float4*)z)[i] = v;
}

// ---------------------------------------------------------------------------
// z[dst[e]] += h[src[e]]   — one wave per edge (32 lanes x float4 = 128 floats)
// Hardware global_atomic_add_f32, one base address + immediate offsets.
// ---------------------------------------------------------------------------
__global__ __launch_bounds__(256) void gin_scatter(const float* __restrict__ h,
                                                   const int* __restrict__ src,
                                                   const int* __restrict__ dst,
                                                   float* __restrict__ z,
                                                   int nedges) {
    int idx = blockIdx.x * blockDim.x + threadIdx.x;
    int e = idx >> 5;
    if (e >= nedges) return;
    int q = (idx & 31) << 2;                       // column 0..124 step 4
    const float4 v = *(const float4*)(h + (size_t)src[e] * DD + q);
    float* zp = z + (size_t)dst[e] * DD + q;
    asm volatile(
        "global_atomic_add_f32 %0, %1, off\n\t"
        "global_atomic_add_f32 %0, %2, off offset:4\n\t"
        "global_atomic_add_f32 %0, %3, off offset:8\n\t"
        "global_atomic_add_f32 %0, %4, off offset:12"
        :: "v"(zp), "v"(v.x), "v"(v.y), "v"(v.z), "v"(v.w) : "memory");
}

// ---------------------------------------------------------------------------
// Fused MLP:  out = ReLU(z @ W1 + b1) @ W2 + b2
// One wave per 16-row tile; V_WMMA_F32_16X16X4_F32 full-precision path.
// All-int indexing: per-step addresses are base + compile-time immediate, so
// loads pipeline. Two N-tiles per iteration -> two independent WMMA chains
// sharing the A fragments.
// ---------------------------------------------------------------------------
__global__ __launch_bounds__(256, 1) void gin_mlp(const float* __restrict__ z,
                                                  const float* __restrict__ W1p,
                                                  const float* __restrict__ b1,
                                                  const float* __restrict__ W2p,
                                                  const float* __restrict__ b2,
                                                  float* __restrict__ out, int n) {
    __shared__ float hidS[8][16 * HROW];           // 67.6 KB of the 320 KB WGP LDS

    const int lane  = threadIdx.x & 31;
    const int wave  = threadIdx.x >> 5;
    const int ln    = lane & 15;                   // N index / A-row index
    const int khoff = lane >> 4;                   // K-pair sub-offset: 0 or 1
    const int rh    = khoff << 3;                  // C/D row-half offset: 0 or 8
    const int m0    = blockIdx.x * 128 + wave * 16;

    int mrow = m0 + ln;
    if (mrow > n - 1) mrow = n - 1;                // clamp loads; stores are guarded
    const float* zrow = z + (size_t)mrow * DD;
    float* hw = &hidS[wave][0];

    const float* az = zrow + 2 * khoff;            // A base, stage 1 (global)
    const float* aw = hw + ln * HROW + 2 * khoff;  // A base, stage 2 (LDS)

    // ---- stage 1: hid = ReLU(z @ W1 + b1), staged into LDS ----
    for (int nt = 0; nt < 8; nt += 2) {
        const int n0 = nt * 16;
        const float* Wb0 = W1p + 2 * (khoff * DD + n0 + ln);        // column tile n0
        const float* Wb1 = W1p + 2 * (khoff * DD + n0 + 16 + ln);   // column tile n0+16
        v8f acc0 = {}, acc1 = {};
        #pragma unroll
        for (int k0 = 0; k0 < DD; k0 += 4) {
            v2f a  = *(const v2f*)(az + k0);       // immediate offset k0*4
            v2f g0 = *(const v2f*)(Wb0 + k0 * DD); // immediate offset k0*DD*4
            v2f g1 = *(const v2f*)(Wb1 + k0 * DD);
            acc0 = __builtin_amdgcn_wmma_f32_16x16x4_f32(false, a, false, g0,
                                                         (short)0, acc0, false, false);
            acc1 = __builtin_amdgcn_wmma_f32_16x16x4_f32(false, a, false, g1,
                                                         (short)0, acc1, false, false);
        }
        const float bias0 = b1[n0 + ln];
        const float bias1 = b1[n0 + 16 + ln];
        #pragma unroll
        for (int j = 0; j < 8; ++j) {
            hw[(j + rh) * HROW + n0 + ln]      = fmaxf(acc0[j] + bias0, 0.0f);
            hw[(j + rh) * HROW + n0 + 16 + ln] = fmaxf(acc1[j] + bias1, 0.0f);
        }
    }

    __syncthreads();

    // ---- stage 2: out = hid @ W2 + b2 ----
    for (int nt = 0; nt < 8; nt += 2) {
        const int n0 = nt * 16;
        const float* Wb0 = W2p + 2 * (khoff * DD + n0 + ln);
        const float* Wb1 = W2p + 2 * (khoff * DD + n0 + 16 + ln);
        v8f acc0 = {}, acc1 = {};
        #pragma unroll
        for (int k0 = 0; k0 < DD; k0 += 4) {
            v2f a  = *(const v2f*)(aw + k0);       // ds_load_b64, immediate offset
            v2f g0 = *(const v2f*)(Wb0 + k0 * DD);
            v2f g1 = *(const v2f*)(Wb1 + k0 * DD);
            acc0 = __builtin_amdgcn_wmma_f32_16x16x4_f32(false, a, false, g0,
                                                         (short)0, acc0, false, false);
            acc1 = __builtin_amdgcn_wmma_f32_16x16x4_f32(false, a, false, g1,
                                                         (short)0, acc1, false, false);
        }
        const float bias0 = b2[n0 + ln];
        const float bias1 = b2[n0 + 16 + ln];
        #pragma unroll
        for (int j = 0; j < 8; ++j) {
            const int r = m0 + j + rh;
            if (r < n) {
                out[(size_t)r * DD + n0 + ln]      = acc0[j] + bias0;
                out[(size_t)r * DD + n0 + 16 + ln] = acc1[j] + bias1;
            }
        }
    }
}

// ---------------------------------------------------------------------------
extern "C" void kernel_launch(void* const* d_in, const int* in_sizes, int n_in,
                              void* d_out, int out_size, void* d_ws, size_t ws_size,
                              hipStream_t stream) {
    const float* h   = (const float*)d_in[0];
    const int*   src = (const int*)  d_in[1];
    const int*   dst = (const int*)  d_in[2];
    const float* W1  = (const float*)d_in[3];
    const float* b1  = (const float*)d_in[4];
    const float* W2  = (const float*)d_in[5];
    const float* b2  = (const float*)d_in[6];
    const float* eps = (const float*)d_in[7];

    const int n = in_sizes[0] / DD;    // 50000 nodes
    const int e = in_sizes[1];         // 800000 edges
    const int L = in_sizes[7];         // 3 layers

    // workspace layout: z | W1p | W2p
    float* z   = (float*)d_ws;                       // [n, DD]
    float* W1p = z + (size_t)n * DD;                 // [L, DD/2, DD, 2]
    float* W2p = W1p + (size_t)L * DD * DD;
    float* out = (float*)d_out;                      // [n, DD] updated in place

    const int n4         = n * DD / 4;
    const int initBlocks = (n4 + 255) / 256;
    const long scatThr   = (long)e * 32;
    const int scatBlocks = (int)((scatThr + 255) / 256);
    const int mlpBlocks  = (n + 127) / 128;
    const int pairs      = L * DD * DD / 2;
    const int rpBlocks   = (pairs + 255) / 256;

    gin_repack<<<rpBlocks, 256, 0, stream>>>(W1, W1p, pairs);
    gin_repack<<<rpBlocks, 256, 0, stream>>>(W2, W2p, pairs);

    for (int l = 0; l < L; ++l) {
        const float* hin = (l == 0) ? h : out;
        gin_init_z<<<initBlocks, 256, 0, stream>>>(hin, eps, l, z, n4);
        gin_scatter<<<scatBlocks, 256, 0, stream>>>(hin, src, dst, z, e);
        gin_mlp<<<mlpBlocks, 256, 0, stream>>>(z, W1p + (size_t)l * DD * DD, b1 + l * DD,
                                               W2p + (size_t)l * DD * DD, b2 + l * DD, out, n);
    }
}